// GCN_12713103196577
// MI455X (gfx1250) — compile-verified
//
#include <hip/hip_runtime.h>
#include <hip/hip_bf16.h>

typedef __attribute__((ext_vector_type(2))) float v2f;
typedef __attribute__((ext_vector_type(8))) float v8f;

#define FOUT 128

// ---------------- fill with zeros ----------------
__global__ void gcn_fill0(float* __restrict__ p, long long n) {
    long long i = (long long)blockIdx.x * blockDim.x + threadIdx.x;
    if (i < n) p[i] = 0.0f;
}

// ---------------- degree accumulation (f32 atomics, exact for counts < 2^24) ----
__global__ void gcn_degree(const int* __restrict__ src, const int* __restrict__ dst,
                           float* __restrict__ odeg, float* __restrict__ ideg, int n_edges) {
    int e = blockIdx.x * blockDim.x + threadIdx.x;
    if (e >= n_edges) return;
    atomicAdd(&odeg[src[e]], 1.0f);
    atomicAdd(&ideg[dst[e]], 1.0f);
}

// deg -> where(deg>0, deg, 1)^-0.5, applied in place over both norm arrays (2N contiguous)
__global__ void gcn_norm(float* __restrict__ deg, int n2) {
    int i = blockIdx.x * blockDim.x + threadIdx.x;
    if (i >= n2) return;
    float d = deg[i];
    deg[i] = (d > 0.0f) ? rsqrtf(d) : 1.0f;
}

// ---------------- layer 1: fin=3, fout=64, scalar (38 MFLOP, negligible) -------
__global__ void gcn_gemm1(const float* __restrict__ x, const float* __restrict__ onorm,
                          const float* __restrict__ W, float* __restrict__ h, int n_nodes) {
    int i = blockIdx.x * blockDim.x + threadIdx.x;
    if (i >= n_nodes * 64) return;
    int n = i >> 6, j = i & 63;
    float s = onorm[n];
    float v = x[n * 3 + 0] * W[j] + x[n * 3 + 1] * W[64 + j] + x[n * 3 + 2] * W[128 + j];
    h[i] = v * s;
}

// ---------------- WMMA f32 GEMM: h[16 x 128] = (x*onorm)[16 x fin] @ W[fin x 128]
// Block = 256 threads = 8 waves; wave w owns output cols [16w, 16w+16).
// A frag (16x4 f32): lanes 0-15 -> K = k,k+1 ; lanes 16-31 -> K = k+2,k+3 (ISA 7.12.2)
// C/D (16x16 f32): VGPR r -> M = r (lanes 0-15) / r+8 (lanes 16-31), N = lane%16
__global__ __launch_bounds__(256)
void gcn_gemm_wmma(const float* __restrict__ x, const float* __restrict__ onorm,
                   const float* __restrict__ W, float* __restrict__ h,
                   int n_nodes, int fin) {
    __shared__ float xs[16 * 129];            // fin <= 128, stride fin+1 avoids bank conflicts
    const int tid  = threadIdx.x;
    const int wave = tid >> 5;
    const int lane = tid & 31;
    const int row0 = blockIdx.x * 16;
    const int stride = fin + 1;

    // stage scaled x tile into LDS
    for (int i = tid; i < 16 * fin; i += 256) {
        int r = i / fin;
        int k = i - r * fin;
        int row = row0 + r;
        float v = 0.0f;
        if (row < n_nodes) v = x[(long long)row * fin + k] * onorm[row];
        xs[r * stride + k] = v;
    }
    __syncthreads();

    const int m     = lane & 15;
    const int khalf = (lane >> 4) << 1;       // 0 or 2
    const int col   = wave * 16 + m;

    v8f acc = {0.0f, 0.0f, 0.0f, 0.0f, 0.0f, 0.0f, 0.0f, 0.0f};
    for (int k = 0; k < fin; k += 4) {
        int kk = k + khalf;
        v2f a, b;
        a.x = xs[m * stride + kk];
        a.y = xs[m * stride + kk + 1];
        b.x = W[kk * FOUT + col];
        b.y = W[(kk + 1) * FOUT + col];
        // 8 args: (neg_a, A, neg_b, B, c_mod, C, reuse_a, reuse_b)
        acc = __builtin_amdgcn_wmma_f32_16x16x4_f32(false, a, false, b, (short)0, acc,
                                                    false, false);
    }

    const int rbase = (lane >> 4) << 3;       // 0 or 8
    #pragma unroll
    for (int r = 0; r < 8; ++r) {
        int row = row0 + rbase + r;
        if (row < n_nodes) h[(long long)row * FOUT + col] = acc[r];
    }
}

// ---------------- edge scatter: m[dst] += h[src]; (F/4) lanes per edge, float4 --
__global__ void gcn_scatter(const float* __restrict__ h, const int* __restrict__ src,
                            const int* __restrict__ dst, float* __restrict__ m,
                            int n_edges, int f) {
    int tpe = f >> 2;                          // threads per edge (16 or 32)
    long long tid = (long long)blockIdx.x * blockDim.x + threadIdx.x;
    long long e = tid / tpe;
    if (e >= n_edges) return;
    int c = (int)(tid - e * tpe) << 2;         // feature offset (multiple of 4)
    int s = src[e], d = dst[e];
    const float4 v = *(const float4*)(h + (long long)s * f + c);
    float* p = m + (long long)d * f + c;
    atomicAdd(p + 0, v.x);
    atomicAdd(p + 1, v.y);
    atomicAdd(p + 2, v.z);
    atomicAdd(p + 3, v.w);
}

// ---------------- relu(m * in_norm + b) in place --------------------------------
__global__ void gcn_bias_relu(float* __restrict__ m, const float* __restrict__ inorm,
                              const float* __restrict__ b, int n_nodes, int f) {
    long long i = (long long)blockIdx.x * blockDim.x + threadIdx.x;
    if (i >= (long long)n_nodes * f) return;
    int n = (int)(i / f);
    int j = (int)(i - (long long)n * f);
    float v = m[i] * inorm[n] + b[j];
    m[i] = v > 0.0f ? v : 0.0f;
}

static inline int blocks_for(long long n, int bs) { return (int)((n + bs - 1) / bs); }

extern "C" void kernel_launch(void* const* d_in, const int* in_sizes, int n_in,
                              void* d_out, int out_size, void* d_ws, size_t ws_size,
                              hipStream_t stream) {
    const float* feat = (const float*)d_in[0];
    const int*   ei   = (const int*)d_in[1];
    const float* W1 = (const float*)d_in[2]; const float* b1 = (const float*)d_in[3];
    const float* W2 = (const float*)d_in[4]; const float* b2 = (const float*)d_in[5];
    const float* W3 = (const float*)d_in[6]; const float* b3 = (const float*)d_in[7];
    const float* W4 = (const float*)d_in[8]; const float* b4 = (const float*)d_in[9];

    const int n_nodes = in_sizes[0] / 3;
    const int n_edges = in_sizes[1] / 2;
    const int* src = ei;
    const int* dst = ei + n_edges;

    float* wsf   = (float*)d_ws;
    float* onorm = wsf;                                   // N
    float* inorm = wsf + n_nodes;                         // N
    float* B0    = wsf + 2 * (size_t)n_nodes;             // N x 128 (h buffer)
    float* B1    = B0 + (size_t)n_nodes * FOUT;           // N x 128 (m / x buffer)
    float* out   = (float*)d_out;

    const int BS = 256;
    const int gemm_blocks = (n_nodes + 15) / 16;

    // degrees -> norms
    gcn_fill0<<<blocks_for(2LL * n_nodes, BS), BS, 0, stream>>>(onorm, 2LL * n_nodes);
    gcn_degree<<<blocks_for(n_edges, BS), BS, 0, stream>>>(src, dst, onorm, inorm, n_edges);
    gcn_norm<<<blocks_for(2LL * n_nodes, BS), BS, 0, stream>>>(onorm, 2 * n_nodes);

    // ---- layer 1: fin=3 -> fout=64 ----
    gcn_gemm1<<<blocks_for((long long)n_nodes * 64, BS), BS, 0, stream>>>(feat, onorm, W1, B0, n_nodes);
    gcn_fill0<<<blocks_for((long long)n_nodes * 64, BS), BS, 0, stream>>>(B1, (long long)n_nodes * 64);
    gcn_scatter<<<blocks_for((long long)n_edges * 16, BS), BS, 0, stream>>>(B0, src, dst, B1, n_edges, 64);
    gcn_bias_relu<<<blocks_for((long long)n_nodes * 64, BS), BS, 0, stream>>>(B1, inorm, b1, n_nodes, 64);

    // ---- layer 2: fin=64 -> 128 (WMMA) ----
    gcn_gemm_wmma<<<gemm_blocks, BS, 0, stream>>>(B1, onorm, W2, B0, n_nodes, 64);
    gcn_fill0<<<blocks_for((long long)n_nodes * FOUT, BS), BS, 0, stream>>>(B1, (long long)n_nodes * FOUT);
    gcn_scatter<<<blocks_for((long long)n_edges * 32, BS), BS, 0, stream>>>(B0, src, dst, B1, n_edges, FOUT);
    gcn_bias_relu<<<blocks_for((long long)n_nodes * FOUT, BS), BS, 0, stream>>>(B1, inorm, b2, n_nodes, FOUT);

    // ---- layer 3: fin=128 -> 128 (WMMA) ----
    gcn_gemm_wmma<<<gemm_blocks, BS, 0, stream>>>(B1, onorm, W3, B0, n_nodes, FOUT);
    gcn_fill0<<<blocks_for((long long)n_nodes * FOUT, BS), BS, 0, stream>>>(B1, (long long)n_nodes * FOUT);
    gcn_scatter<<<blocks_for((long long)n_edges * 32, BS), BS, 0, stream>>>(B0, src, dst, B1, n_edges, FOUT);
    gcn_bias_relu<<<blocks_for((long long)n_nodes * FOUT, BS), BS, 0, stream>>>(B1, inorm, b3, n_nodes, FOUT);

    // ---- layer 4: fin=128 -> 128 (WMMA), scatter straight into d_out ----
    gcn_gemm_wmma<<<gemm_blocks, BS, 0, stream>>>(B1, onorm, W4, B0, n_nodes, FOUT);
    gcn_fill0<<<blocks_for((long long)n_nodes * FOUT, BS), BS, 0, stream>>>(out, (long long)n_nodes * FOUT);
    gcn_scatter<<<blocks_for((long long)n_edges * 32, BS), BS, 0, stream>>>(B0, src, dst, out, n_edges, FOUT);
    gcn_bias_relu<<<blocks_for((long long)n_nodes * FOUT, BS), BS, 0, stream>>>(out, inorm, b4, n_nodes, FOUT);
}